// HRQuantizeEMAResetV2_12678743458267
// MI455X (gfx1250) — compile-verified
//
#include <hip/hip_runtime.h>
#include <hip/hip_bf16.h>

typedef __attribute__((ext_vector_type(2))) float v2f;
typedef __attribute__((ext_vector_type(8))) float v8f;

#define N_B   32
#define C_DIM 512
#define T_DIM 2048
#define K_CB  1024

// ---------------------------------------------------------------- init
__global__ void var_init_kernel(const float* __restrict__ x,
                                float* __restrict__ residual,
                                float* __restrict__ fhat, int n) {
  for (int e = blockIdx.x * blockDim.x + threadIdx.x; e < n;
       e += gridDim.x * blockDim.x) {
    residual[e] = x[e];
    fhat[e] = 0.f;
  }
}

// ------------------------------------------------------- codebook norms
__global__ void var_cbnorm_kernel(const float* __restrict__ cb,
                                  float* __restrict__ cnorm) {
  int wave = threadIdx.x >> 5, lane = threadIdx.x & 31;
  int r = blockIdx.x * 8 + wave;
  const float* row = cb + (size_t)r * C_DIM;
  float s = 0.f;
  for (int j = lane; j < C_DIM; j += 32) { float v = row[j]; s += v * v; }
  #pragma unroll
  for (int m = 16; m >= 1; m >>= 1) s += __shfl_xor(s, m, 32);
  if (lane == 0) cnorm[r] = s;
}

// ----------------------------------------------------------- downsample
__global__ void var_downsample_kernel(const float* __restrict__ residual,
                                      float* __restrict__ rd, int s, int Td,
                                      int total) {
  float inv = 1.f / (float)s;
  for (int e = blockIdx.x * blockDim.x + threadIdx.x; e < total;
       e += gridDim.x * blockDim.x) {
    int td = e % Td;
    int c  = (e / Td) % C_DIM;
    int n  = e / (Td * C_DIM);
    const float* p = residual + ((size_t)n * C_DIM + c) * T_DIM + (size_t)td * s;
    float acc = 0.f;
    for (int j = 0; j < s; ++j) acc += p[j];
    rd[e] = acc * inv;
  }
}

// --------------------------------------------- WMMA nearest-codeword GEMM
// block: 256 thr (8 waves); 32 rows (2 M-subtiles) vs all 1024 codewords.
// A tile stored as two 16-row planes with lane-pair-contiguous K pairs so a
// fragment is one conflict-free ds_load_b64 into an even VGPR pair.
__global__ void __launch_bounds__(256)
var_quantize_kernel(const float* __restrict__ rd, const float* __restrict__ cb,
                    const float* __restrict__ cnorm, int* __restrict__ idx_out,
                    float* __restrict__ idxf_out, int Td) {
  extern __shared__ float smem[];
  float* a_lds = smem;                       // 2 planes * 8192 floats
  float* s_val = smem + 2 * 8192;            // [8][32]
  int*   s_idx = (int*)(s_val + 8 * 32);     // [8][32]

  int m0 = blockIdx.x * 32;
  int n  = m0 / Td, tb = m0 % Td;            // Td % 32 == 0 for all scales
  const float* base = rd + (size_t)n * C_DIM * Td + tb;

  for (int j = threadIdx.x; j < C_DIM * 32; j += 256) {
    int row = j & 31, c = j >> 5;
    // plane = row/16, pair = c/2, slot = (row%16)*2 + (c&1)
    a_lds[(row >> 4) * 8192 + (c >> 1) * 32 + (row & 15) * 2 + (c & 1)] =
        base[(size_t)c * Td + row];
  }
  __syncthreads();

  int wave = threadIdx.x >> 5, lane = threadIdx.x & 31;
  int half = lane >> 4, lcol = lane & 15;
  int koff = half ? 2 : 0;

  float bestv[2][8];
  int   besti[2][8];
  #pragma unroll
  for (int mt = 0; mt < 2; ++mt)
    #pragma unroll
    for (int v = 0; v < 8; ++v) { bestv[mt][v] = 3.4e38f; besti[mt][v] = 0; }

  for (int cwt = 0; cwt < 8; ++cwt) {
    int cw0 = (wave * 8 + cwt) * 16;
    v8f acc0 = {}; v8f acc1 = {};
    const float* cbp = cb + (size_t)(cw0 + lcol) * C_DIM + koff;
    if (cwt < 7)  // warm next tile's B stream (global_prefetch_b8)
      __builtin_prefetch(cbp + 16 * C_DIM, 0, 1);
    #pragma unroll 4
    for (int c0 = 0; c0 < C_DIM; c0 += 4) {
      v2f b = *(const v2f*)(cbp + c0);             // one global_load_b64
      int pA = (c0 + koff) >> 1;                   // K pair index
      v2f a0 = *(const v2f*)&a_lds[pA * 32 + lcol * 2];
      v2f a1 = *(const v2f*)&a_lds[8192 + pA * 32 + lcol * 2];
      acc0 = __builtin_amdgcn_wmma_f32_16x16x4_f32(false, a0, false, b,
                                                   (short)0, acc0, false, false);
      acc1 = __builtin_amdgcn_wmma_f32_16x16x4_f32(false, a1, false, b,
                                                   (short)0, acc1, false, false);
    }
    float cn = cnorm[cw0 + lcol];
    int   ci = cw0 + lcol;
    #pragma unroll
    for (int v = 0; v < 8; ++v) {
      float d0 = cn - 2.f * acc0[v];
      float d1 = cn - 2.f * acc1[v];
      if (d0 < bestv[0][v]) { bestv[0][v] = d0; besti[0][v] = ci; }
      if (d1 < bestv[1][v]) { bestv[1][v] = d1; besti[1][v] = ci; }
    }
  }

  // butterfly argmin across each 16-lane half (D-layout half), tie -> low idx
  #pragma unroll
  for (int mt = 0; mt < 2; ++mt)
    #pragma unroll
    for (int v = 0; v < 8; ++v) {
      float bv = bestv[mt][v];
      int   bi = besti[mt][v];
      #pragma unroll
      for (int msk = 1; msk <= 8; msk <<= 1) {
        float ov = __shfl_xor(bv, msk, 32);
        int   oi = __shfl_xor(bi, msk, 32);
        if (ov < bv || (ov == bv && oi < bi)) { bv = ov; bi = oi; }
      }
      if (lcol == 0) {
        int srow = mt * 16 + v + 8 * half;
        s_val[wave * 32 + srow] = bv;
        s_idx[wave * 32 + srow] = bi;
      }
    }
  __syncthreads();

  if (threadIdx.x < 32) {
    int row = threadIdx.x;
    float bv = s_val[row];
    int   bi = s_idx[row];
    for (int w = 1; w < 8; ++w) {
      float ov = s_val[w * 32 + row];
      int   oi = s_idx[w * 32 + row];
      if (ov < bv || (ov == bv && oi < bi)) { bv = ov; bi = oi; }
    }
    idx_out[m0 + row]  = bi;
    idxf_out[m0 + row] = (float)bi;
  }
}

// -------------------------------------------- gather + linear upsample
__global__ void var_gather_up_kernel(const int* __restrict__ idx,
                                     const float* __restrict__ cb,
                                     float* __restrict__ up, int s, int Td,
                                     int total) {
  float invs = 1.f / (float)s;
  for (int e = blockIdx.x * blockDim.x + threadIdx.x; e < total;
       e += gridDim.x * blockDim.x) {
    int t = e % T_DIM;
    int c = (e / T_DIM) % C_DIM;
    int n = e / (T_DIM * C_DIM);
    float coord = fmaxf(((float)t + 0.5f) * invs - 0.5f, 0.f);
    int i0 = (int)floorf(coord);
    if (i0 > Td - 1) i0 = Td - 1;
    int i1 = min(i0 + 1, Td - 1);
    float w = coord - (float)i0;
    int k0 = idx[n * Td + i0], k1 = idx[n * Td + i1];
    up[e] = cb[(size_t)k0 * C_DIM + c] * (1.f - w) +
            cb[(size_t)k1 * C_DIM + c] * w;
  }
}

// ----------------------- WMMA conv1d(k=3) + Phi blend + residual/f_hat
// block: 16 couts x 128 t; W tile im2col'ed in LDS (k = tap*512 + cin),
// pair-contiguous so an A fragment is one conflict-free ds_load_b64.
__global__ void __launch_bounds__(256)
var_conv_phi_kernel(const float* __restrict__ up, const float* __restrict__ w,
                    const float* __restrict__ bias,
                    float* __restrict__ residual, float* __restrict__ fhat) {
  extern __shared__ float w_lds[];   // 768 pairs * 32 floats
  int t_blk = blockIdx.x * 128;
  int cout0 = blockIdx.y * 16;
  int n     = blockIdx.z;

  for (int j = threadIdx.x; j < 1536 * 16; j += 256) {
    int m = j & 15, kk = j >> 4;
    int tap = kk >> 9, cin = kk & 511;
    w_lds[(kk >> 1) * 32 + m * 2 + (kk & 1)] =
        w[(size_t)(cout0 + m) * 1536 + cin * 3 + tap];
  }
  __syncthreads();

  int wave = threadIdx.x >> 5, lane = threadIdx.x & 31;
  int half = lane >> 4, lcol = lane & 15;
  int koff = half ? 2 : 0;
  int t0 = t_blk + wave * 16;

  v8f acc = {};
  const float* upn = up + (size_t)n * C_DIM * T_DIM;
  #pragma unroll
  for (int tap = 0; tap < 3; ++tap) {
    int tg = t0 + lcol + tap - 1;
    float msk = (tg >= 0 && tg < T_DIM) ? 1.f : 0.f;   // zero-pad, branchless
    int tgc = min(max(tg, 0), T_DIM - 1);
    #pragma unroll 4
    for (int c0 = 0; c0 < C_DIM; c0 += 4) {
      int cB = c0 + koff;
      const float* p = upn + (size_t)cB * T_DIM + tgc;
      v2f b; b.x = p[0] * msk; b.y = p[T_DIM] * msk;
      int pA = (tap * C_DIM + cB) >> 1;
      v2f a = *(const v2f*)&w_lds[pA * 32 + lcol * 2];
      acc = __builtin_amdgcn_wmma_f32_16x16x4_f32(false, a, false, b,
                                                  (short)0, acc, false, false);
    }
  }

  int t = t0 + lcol;
  #pragma unroll
  for (int v = 0; v < 8; ++v) {
    int cout = cout0 + v + 8 * half;
    size_t addr = ((size_t)n * C_DIM + cout) * T_DIM + t;
    float phi = up[addr] * 0.5f + (acc[v] + bias[cout]) * 0.5f;  // RESI = 0.5
    residual[addr] -= phi;
    fhat[addr]     += phi;
  }
}

// ---------------------------------------------------------------- launch
extern "C" void kernel_launch(void* const* d_in, const int* in_sizes, int n_in,
                              void* d_out, int out_size, void* d_ws,
                              size_t ws_size, hipStream_t stream) {
  const float* x     = (const float*)d_in[0];
  const float* cb    = (const float*)d_in[1];
  const float* phi_w = (const float*)d_in[2];
  const float* phi_b = (const float*)d_in[3];
  float* out = (float*)d_out;

  const size_t nct = (size_t)N_B * C_DIM * T_DIM;   // 33,554,432
  float* residual = (float*)d_ws;
  float* up       = residual + nct;
  float* rd       = up + nct;                        // max N*C*1024
  int*   idxbuf   = (int*)(rd + nct / 2);
  float* cnorm    = (float*)(idxbuf + N_B * T_DIM);

  // d_out layout: idx(s=1..8) as float, then f_hat
  float* fhat = out + N_B * (2048 + 1024 + 512 + 256);   // + 122880

  var_init_kernel<<<4096, 256, 0, stream>>>(x, residual, fhat, (int)nct);
  var_cbnorm_kernel<<<K_CB / 8, 256, 0, stream>>>(cb, cnorm);

  const int scales[4] = {1, 2, 4, 8};
  int off = 0;
  for (int i = 0; i < 4; ++i) {
    int s = scales[i], Td = T_DIM / s;
    const float* src = residual;
    if (s > 1) {
      int total = N_B * C_DIM * Td;
      var_downsample_kernel<<<4096, 256, 0, stream>>>(residual, rd, s, Td, total);
      src = rd;
    }
    int M = N_B * Td;
    size_t q_lds = (size_t)2 * 8192 * 4 + 8 * 32 * 4 + 8 * 32 * 4; // 67584 B
    var_quantize_kernel<<<M / 32, 256, q_lds, stream>>>(src, cb, cnorm, idxbuf,
                                                        out + off, Td);
    var_gather_up_kernel<<<4096, 256, 0, stream>>>(idxbuf, cb, up, s, Td,
                                                   (int)nct);
    var_conv_phi_kernel<<<dim3(T_DIM / 128, C_DIM / 16, N_B), 256,
                          (size_t)1536 * 16 * 4, stream>>>(
        up, phi_w + (size_t)i * C_DIM * C_DIM * 3, phi_b + (size_t)i * C_DIM,
        residual, fhat);
    off += M;
  }
}